// DTMLayer_63531156242953
// MI455X (gfx1250) — compile-verified
//
#include <hip/hip_runtime.h>
#include <math.h>

typedef __attribute__((ext_vector_type(2))) float    v2f;
typedef __attribute__((ext_vector_type(8))) float    v8f;
typedef __attribute__((ext_vector_type(4))) unsigned u32x4;
typedef __attribute__((ext_vector_type(8))) int      i32x8;
typedef __attribute__((ext_vector_type(4))) int      i32x4;

namespace {
constexpr int   kBatch      = 4;
constexpr int   kM          = 4096;    // points per batch
constexpr int   kN          = 10201;   // grid points (101*101)
constexpr int   kK          = 1229;    // ceil(0.3*4096)
constexpr float kWB         = 1228.8f; // 0.3*4096
constexpr int   kRows       = 16;      // grid rows per workgroup (WMMA M)
constexpr int   kThreads    = 256;     // 8 waves of 32
constexpr int   kWaves      = 8;
constexpr int   kPtsPerWave = kM / kWaves;      // 512
constexpr int   kGroups     = kPtsPerWave / 16; // 32 WMMA tiles per wave per sweep
constexpr int   kHistRows   = kRows + 1;        // +1 trash row -> branchless updates
}

// One workgroup = (batch b, 16 consecutive grid rows).
// d^2(grid_i, pt_j) computed EXACTLY as one K=4 f32 WMMA per 16x16 tile:
//   A[i,:] = { gx, gy, |g|^2, 1 } ,  B[:,j] = { -2px, -2py, 1, |p|^2 }
// k-th smallest d^2 found by 4-pass radix select on IEEE bits (d^2 >= 0 so
// uint order == float order); distances recomputed per pass via WMMA.
__global__ __launch_bounds__(kThreads)
void dtm_wmma_kernel(const float* __restrict__ x,
                     const float* __restrict__ grid,
                     float* __restrict__ out)
{
    __shared__ __align__(16) float xs[kM * 2];   // staged x[b], 32 KB
    __shared__ unsigned hist[kHistRows * 256];   // 17 KB (row 16 = trash bin)
    __shared__ unsigned pfx[kRows];              // resolved high bits of T
    __shared__ unsigned krem[kRows];             // remaining rank in bucket
    __shared__ float    rowSumW[kRows * kWaves];
    __shared__ unsigned rowCntW[kRows * kWaves];

    const int tid  = threadIdx.x;
    const int wave = tid >> 5;
    const int lane = tid & 31;
    const int half = lane >> 4;   // 0: lanes 0-15, 1: lanes 16-31
    const int l16  = lane & 15;
    const int b    = blockIdx.y;
    const int rowBase = blockIdx.x * kRows;

    // ---- stage x[b] (8192 f32 = 32 KB) into LDS via one TDM DMA ----
    const float* xbase = x + (size_t)b * kM * 2;
#if __has_builtin(__builtin_amdgcn_tensor_load_to_lds) && \
    __has_builtin(__builtin_amdgcn_s_wait_tensorcnt)
    if (wave == 0) {
        const unsigned long long ga = (unsigned long long)(uintptr_t)xbase;
        const unsigned ldsa = (unsigned)(uintptr_t)xs;  // low 32b = LDS offset
        u32x4 g0;
        g0[0] = 1u;                                        // count=1 (valid D#)
        g0[1] = ldsa;                                      // lds_addr
        g0[2] = (unsigned)ga;                              // global_addr[31:0]
        g0[3] = (unsigned)((ga >> 32) & 0x01FFFFFFu)       // global_addr[56:32]
              | (2u << 30);                                // type=2 ("image")
        i32x8 g1;
        g1[0] = (int)(2u << 16);              // data_size=4B, wg_mask=0
        g1[1] = (int)(8192u << 16);           // tensor_dim0[15:0]=8192
        g1[2] = (int)((8192u >> 16) | (1u << 16)); // dim0[31:16] | tensor_dim1=1
        g1[3] = (int)(8192u << 16);           // dim1[31:16]=0 | tile_dim0=8192
        g1[4] = 1;                            // tile_dim1=1, tile_dim2=0
        g1[5] = 8192;                         // tensor_dim0_stride
        g1[6] = 0;
        g1[7] = 0;
        i32x4 gz4 = {0, 0, 0, 0};
        i32x8 gz8 = {0, 0, 0, 0, 0, 0, 0, 0};
        // 6-arg form (this toolchain): (g0, g1, g2, g3, extra, cpol)
        __builtin_amdgcn_tensor_load_to_lds(g0, g1, gz4, gz4, gz8, 0);
        __builtin_amdgcn_s_wait_tensorcnt(0);              // s_wait_tensorcnt 0
    }
#else
    {
        const float4* xg = reinterpret_cast<const float4*>(xbase);
        float4* xl = reinterpret_cast<float4*>(xs);
        __builtin_prefetch(xg + tid, 0, 1);
#pragma unroll
        for (int i = 0; i < (kM * 2 / 4) / kThreads; ++i)
            xl[tid + i * kThreads] = xg[tid + i * kThreads];
    }
#endif

    if (tid < kRows) { pfx[tid] = 0u; krem[tid] = (unsigned)kK; }

    // ---- WMMA A operand (constant per workgroup): 16x4 f32 ----
    // VGPR0 = lanes0-15:K0, lanes16-31:K2 ; VGPR1 = lanes0-15:K1, lanes16-31:K3
    int rA = rowBase + l16;
    rA = (rA < kN) ? rA : (kN - 1);              // clamp tail tile
    const float gx = grid[2 * rA + 0];
    const float gy = grid[2 * rA + 1];
    v2f Aop;
    Aop.x = half ? (gx * gx + gy * gy) : gx;     // K2=|g|^2 : K0=gx
    Aop.y = half ? 1.0f : gy;                    // K3=1     : K1=gy

    __syncthreads();

    const float2* xp = reinterpret_cast<const float2*>(xs);
    const int mBase = wave * kPtsPerWave + l16;

    // =============== 4-pass radix select on float bits ===============
    unsigned dmask = 0u;
    for (int p = 0; p < 4; ++p) {
        const int shift = 24 - 8 * p;

        for (int i = tid; i < kHistRows * 256; i += kThreads) hist[i] = 0u;
        __syncthreads();

        unsigned pr[8];
#pragma unroll
        for (int v = 0; v < 8; ++v) pr[v] = pfx[v + 8 * half];

        float2 cur = xp[mBase];                 // software-pipelined B loads
#pragma unroll 2
        for (int mg = 0; mg < kGroups; ++mg) {
            const float2 nxt = (mg + 1 < kGroups) ? xp[mBase + (mg + 1) * 16] : cur;
            const float px = cur.x, py = cur.y;
            v2f Bop;
            Bop.x = half ? (-2.0f * py) : (-2.0f * px);   // K1 : K0
            Bop.y = half ? (px * px + py * py) : 1.0f;    // K3 : K2
            v8f acc = {};
            acc = __builtin_amdgcn_wmma_f32_16x16x4_f32(
                false, Aop, false, Bop, (short)0, acc, false, false);
#pragma unroll
            for (int v = 0; v < 8; ++v) {
                const float d2 = fmaxf(acc[v], 0.0f);
                const unsigned u = __float_as_uint(d2);
                const bool match = (u & dmask) == pr[v];
                // branchless: non-matching lanes hit the trash row (row 16)
                const unsigned base =
                    match ? (unsigned)((v + 8 * half) * 256) : (unsigned)(kRows * 256);
                const unsigned dgt = (u >> shift) & 255u;
                atomicAdd(&hist[base + dgt], 1u);         // ds_add_u32, full EXEC
            }
            cur = nxt;
        }
        __syncthreads();

        if (tid < kRows) {                      // pick digit holding rank krem
            unsigned kr = krem[tid], run = 0, dsel = 255u;
            const unsigned* h = &hist[tid * 256];
            for (int d = 0; d < 256; ++d) {
                const unsigned cd = h[d];
                if (run + cd >= kr) { dsel = (unsigned)d; break; }
                run += cd;
            }
            pfx[tid] |= dsel << shift;
            krem[tid] = kr - run;
        }
        dmask |= 0xFFu << shift;
        __syncthreads();
    }

    // =============== final sweep: sum of d^2 strictly below T ===============
    unsigned Tb[8];
#pragma unroll
    for (int v = 0; v < 8; ++v) Tb[v] = pfx[v + 8 * half];

    float    ps[8] = {0.f, 0.f, 0.f, 0.f, 0.f, 0.f, 0.f, 0.f};
    unsigned pc[8] = {0u, 0u, 0u, 0u, 0u, 0u, 0u, 0u};

    float2 cur = xp[mBase];
#pragma unroll 2
    for (int mg = 0; mg < kGroups; ++mg) {
        const float2 nxt = (mg + 1 < kGroups) ? xp[mBase + (mg + 1) * 16] : cur;
        const float px = cur.x, py = cur.y;
        v2f Bop;
        Bop.x = half ? (-2.0f * py) : (-2.0f * px);
        Bop.y = half ? (px * px + py * py) : 1.0f;
        v8f acc = {};
        acc = __builtin_amdgcn_wmma_f32_16x16x4_f32(
            false, Aop, false, Bop, (short)0, acc, false, false);
#pragma unroll
        for (int v = 0; v < 8; ++v) {
            const float d2 = fmaxf(acc[v], 0.0f);
            const unsigned u = __float_as_uint(d2);
            const bool lt = u < Tb[v];                 // uint order == float order
            ps[v] += lt ? d2 : 0.0f;
            pc[v] += lt ? 1u : 0u;
        }
        cur = nxt;
    }

    // reduce across the 16 lanes sharing each row (xor offsets stay in-half)
#pragma unroll
    for (int v = 0; v < 8; ++v) {
#pragma unroll
        for (int off = 8; off > 0; off >>= 1) {
            ps[v] += __shfl_xor(ps[v], off);
            pc[v] += __shfl_xor(pc[v], off);
        }
    }
    if (l16 == 0) {
#pragma unroll
        for (int v = 0; v < 8; ++v) {
            rowSumW[(v + 8 * half) * kWaves + wave] = ps[v];
            rowCntW[(v + 8 * half) * kWaves + wave] = pc[v];
        }
    }
    __syncthreads();

    // =============== DTM aggregation + output ===============
    if (tid < kRows) {
        const int row = rowBase + tid;
        if (row < kN) {
            float s = 0.f; unsigned c = 0u;
#pragma unroll
            for (int w = 0; w < kWaves; ++w) {         // deterministic order
                s += rowSumW[tid * kWaves + w];
                c += rowCntW[tid * kWaves + w];
            }
            const float T = __uint_as_float(pfx[tid]); // k-th smallest d^2
            // cum[k-1] + T*(wb-k) == s + T*(k-c) + T*(wb-k) == s + T*(wb-c)
            const float tot = s + T * (kWB - (float)c);
            out[(size_t)b * kN + row] = sqrtf(tot / kWB);
        }
    }
}

extern "C" void kernel_launch(void* const* d_in, const int* in_sizes, int n_in,
                              void* d_out, int out_size, void* d_ws, size_t ws_size,
                              hipStream_t stream) {
    const float* x    = (const float*)d_in[0];   // [4, 4096, 2] f32
    const float* grid = (const float*)d_in[1];   // [10201, 2]   f32
    float* out        = (float*)d_out;           // [4, 10201]   f32
    dim3 g((kN + kRows - 1) / kRows, kBatch);    // 638 row-tiles x 4 batches
    dtm_wmma_kernel<<<g, kThreads, 0, stream>>>(x, grid, out);
}